// get_loss_84610855731482
// MI455X (gfx1250) — compile-verified
//
#include <hip/hip_runtime.h>
#include <math.h>

typedef float v2f __attribute__((ext_vector_type(2)));
typedef float v4f __attribute__((ext_vector_type(4)));
typedef float v8f __attribute__((ext_vector_type(8)));

#define PD_EPS 1e-6f

__device__ __forceinline__ float wave_reduce_add(float v) {
#pragma unroll
    for (int off = 16; off > 0; off >>= 1)
        v += __shfl_xor(v, off, 32);
    return v;
}

// ---------------------------------------------------------------------------
// Pass 1: per-row focal loss term + label extraction (one wave32 per row).
// p_t = sum(P * onehot) ; label = sum(c * onehot[c]) (exact one-hot).
// ---------------------------------------------------------------------------
__global__ void focal_label_kernel(const float* __restrict__ prob,
                                   const float* __restrict__ gt,
                                   int N, int C,
                                   int* __restrict__ labels,
                                   float* __restrict__ acc) {
    const int lane = threadIdx.x & 31;
    const int wib  = threadIdx.x >> 5;
    const int wid  = (blockIdx.x * blockDim.x + threadIdx.x) >> 5;
    const int nw   = (gridDim.x * blockDim.x) >> 5;

    float wsum = 0.0f;
    for (int row = wid; row < N; row += nw) {
        const float* p = prob + (size_t)row * C;
        const float* g = gt   + (size_t)row * C;
        float pt = 0.0f, lab = 0.0f;
        for (int c = lane; c < C; c += 32) {
            float gv = g[c];
            pt  = fmaf(p[c], gv, pt);
            lab = fmaf((float)c, gv, lab);
        }
        pt  = wave_reduce_add(pt);
        lab = wave_reduce_add(lab);
        if (lane == 0) {
            labels[row] = (int)(lab + 0.5f);
            float om = 1.0f - pt;
            wsum -= om * om * __logf(pt);   // -(1-p_t)^2 * log(p_t)
        }
    }

    __shared__ float sh[8];
    if (lane == 0) sh[wib] = wsum;
    __syncthreads();
    if (threadIdx.x == 0) {
        float s = 0.0f;
        const int nwb = blockDim.x >> 5;
        for (int i = 0; i < nwb; ++i) s += sh[i];
        atomicAdd(acc, s);
    }
}

// ---------------------------------------------------------------------------
// Pass 2: prototype-distance sum. One wave32 handles 16 rows (D = 512).
// diff^2 values are fed into V_WMMA_F32_16X16X4_F32 with an all-ones B
// matrix: with B == 1, C[m][n] accumulates the row-sum of A, so the WMMA
// unit performs the 16-row horizontal reduction.  f32 A layout (16x4):
//   lanes 0-15  supply K=0,1 of row = lane
//   lanes 16-31 supply K=2,3 of row = lane-16
// Summation is order-invariant, so lane-half L covers dims [base+4L,base+4L+3]
// => every lane does aligned 16B loads and all 8 dims per WMMA-pair are
// counted exactly once.  The k-loop is software-pipelined depth-4 (8 b128
// loads in flight per wave) to keep HBM busy without s_wait_loadcnt 0 stalls.
// ---------------------------------------------------------------------------
__device__ __forceinline__ void wmma_accum_chunk(v4f c4, v4f p4,
                                                 const v2f& ones,
                                                 v8f& acc0, v8f& acc1) {
    float d0 = (c4.x - p4.x) + PD_EPS;
    float d1 = (c4.y - p4.y) + PD_EPS;
    float d2 = (c4.z - p4.z) + PD_EPS;
    float d3 = (c4.w - p4.w) + PD_EPS;
    v2f a0; a0.x = d0 * d0; a0.y = d1 * d1;
    v2f a1; a1.x = d2 * d2; a1.y = d3 * d3;
    acc0 = __builtin_amdgcn_wmma_f32_16x16x4_f32(
        false, a0, false, ones, (short)0, acc0, false, false);
    acc1 = __builtin_amdgcn_wmma_f32_16x16x4_f32(
        false, a1, false, ones, (short)0, acc1, false, false);
}

__global__ void dist_wmma_kernel(const float* __restrict__ codes,
                                 const float* __restrict__ proto,
                                 const int* __restrict__ labels,
                                 int N,
                                 float* __restrict__ acc) {
    const int D    = 512;           // multiple of 32 (pipeline stride)
    const int PIPE = 4;             // chunks of 8 floats in flight
    const int lane = threadIdx.x & 31;
    const int wib  = threadIdx.x >> 5;
    const int wpb  = blockDim.x >> 5;
    const int nGroups = N >> 4;

    const v2f ones = {1.0f, 1.0f};
    float wsum = 0.0f;

    for (int grp = blockIdx.x * wpb + wib; grp < nGroups; grp += gridDim.x * wpb) {
        const int row     = (grp << 4) + (lane & 15);
        const int lab     = labels[row];
        const int halfOff = (lane >> 4) << 2;  // 0 for lanes 0-15, 4 for 16-31
        const float* crow = codes + (size_t)row * D + halfOff;
        const float* prow = proto + (size_t)lab * D + halfOff;

        v8f acc0 = {0.f, 0.f, 0.f, 0.f, 0.f, 0.f, 0.f, 0.f};
        v8f acc1 = {0.f, 0.f, 0.f, 0.f, 0.f, 0.f, 0.f, 0.f};

        // Prologue: fill the pipeline.
        v4f cbuf[PIPE], pbuf[PIPE];
#pragma unroll
        for (int i = 0; i < PIPE; ++i) {
            cbuf[i] = *(const v4f*)(crow + 8 * i);
            pbuf[i] = *(const v4f*)(prow + 8 * i);
        }

        // Main loop: issue next iteration's 8 loads, then compute on the
        // previously loaded registers (D multiple of 8*PIPE keeps this exact).
        for (int base = 0; base + 8 * PIPE < D; base += 8 * PIPE) {
            __builtin_prefetch(crow + base + 8 * PIPE + 64, 0, 3);  // WGP-scope
#pragma unroll
            for (int i = 0; i < PIPE; ++i) {
                v4f c4 = cbuf[i];
                v4f p4 = pbuf[i];
                cbuf[i] = *(const v4f*)(crow + base + 8 * PIPE + 8 * i);
                pbuf[i] = *(const v4f*)(prow + base + 8 * PIPE + 8 * i);
                wmma_accum_chunk(c4, p4, ones, acc0, acc1);
            }
        }
        // Epilogue: drain the pipeline.
#pragma unroll
        for (int i = 0; i < PIPE; ++i)
            wmma_accum_chunk(cbuf[i], pbuf[i], ones, acc0, acc1);

        // Columns of C are identical (B all-ones): lanes 0-15 hold rows 0-7 in
        // VGPRs 0-7, lanes 16-31 hold rows 8-15.  sqrt each row sum and add.
        float S = 0.0f;
#pragma unroll
        for (int v = 0; v < 8; ++v) S += sqrtf(acc0[v] + acc1[v]);
        S += __shfl_xor(S, 16, 32);   // rows 0-7 + rows 8-15
        if (lane == 0) wsum += S;
    }

    __shared__ float sh[8];
    if (lane == 0) sh[wib] = wsum;
    __syncthreads();
    if (threadIdx.x == 0) {
        float s = 0.0f;
        for (int i = 0; i < wpb; ++i) s += sh[i];
        atomicAdd(acc, s);
    }
}

__global__ void zero_acc_kernel(float* __restrict__ acc) {
    if (threadIdx.x < 4) acc[threadIdx.x] = 0.0f;
}

__global__ void finalize_kernel(const float* __restrict__ acc,
                                float* __restrict__ out,
                                float invNn, float invNe) {
    // loss = obj_focal/Nn + pred_focal/Ne + node_dist/Nn + edge_dist/Ne
    out[0] = acc[0] * invNn + acc[1] * invNe + acc[2] * invNn + acc[3] * invNe;
}

extern "C" void kernel_launch(void* const* d_in, const int* in_sizes, int n_in,
                              void* d_out, int out_size, void* d_ws, size_t ws_size,
                              hipStream_t stream) {
    const float* node_output = (const float*)d_in[0];
    const float* edge_output = (const float*)d_in[1];
    const float* node_codes  = (const float*)d_in[2];
    const float* edge_codes  = (const float*)d_in[3];
    const float* node_meta   = (const float*)d_in[4];
    const float* edge_meta   = (const float*)d_in[5];
    const float* node_gt     = (const float*)d_in[6];
    const float* edge_gt     = (const float*)d_in[7];

    const int D  = 512;
    const int Nn = in_sizes[2] / D;        // 65536
    const int Ne = in_sizes[3] / D;        // 262144
    const int Cn = in_sizes[0] / Nn;       // 160
    const int Ce = in_sizes[1] / Ne;       // 27

    // Workspace: [0..15] four float accumulators; labels at offset 256.
    float* acc  = (float*)d_ws;
    int*   labN = (int*)((char*)d_ws + 256);
    int*   labE = labN + Nn;

    zero_acc_kernel<<<1, 32, 0, stream>>>(acc);

    focal_label_kernel<<<1024, 256, 0, stream>>>(node_output, node_gt, Nn, Cn, labN, acc + 0);
    focal_label_kernel<<<2048, 256, 0, stream>>>(edge_output, edge_gt, Ne, Ce, labE, acc + 1);

    dist_wmma_kernel<<<512,  256, 0, stream>>>(node_codes, node_meta, labN, Nn, acc + 2);
    dist_wmma_kernel<<<2048, 256, 0, stream>>>(edge_codes, edge_meta, labE, Ne, acc + 3);

    finalize_kernel<<<1, 1, 0, stream>>>(acc, (float*)d_out, 1.0f / Nn, 1.0f / Ne);
}